// SplineBasis_89670327206591
// MI455X (gfx1250) — compile-verified
//
#include <hip/hip_runtime.h>
#include <hip/hip_bf16.h>

#define GRID_SIZE 64
#define KDEG 3
#define NB (GRID_SIZE + KDEG)        // 67 basis functions
#define ROWS_PER_BLOCK 256
#define NTHREADS 256

typedef __attribute__((ext_vector_type(4))) unsigned int v4u;
typedef __attribute__((ext_vector_type(8))) int          v8i;
typedef __attribute__((ext_vector_type(4))) int          v4i;
typedef __attribute__((ext_vector_type(4))) float        f4;

// Clamped uniform knot vector: knot[m] = clamp(-1 + (m-K)*h, -1, 1), h = 2/64
__device__ __forceinline__ float knotf(int m) {
    float t = -1.0f + (float)(m - KDEG) * 0.03125f;
    return fminf(1.0f, fmaxf(-1.0f, t));
}

__global__ __launch_bounds__(NTHREADS)
void spline_basis_kernel(const float* __restrict__ x_in,
                         float* __restrict__ out, int B) {
    __shared__ __align__(16) float tile[ROWS_PER_BLOCK * NB];   // 68,608 B
    const int tid = threadIdx.x;
    const int r0  = blockIdx.x * ROWS_PER_BLOCK;

    // 1) zero-fill the LDS tile with b128 stores
    f4* tile4 = (f4*)tile;
    const int n4 = (ROWS_PER_BLOCK * NB) / 4;                    // 4288, divisible
    f4 zf4 = {0.f, 0.f, 0.f, 0.f};
    for (int i = tid; i < n4; i += NTHREADS) tile4[i] = zf4;
    __syncthreads();

    // 2) de Boor: 4 nonzero cubic basis values for this point, scatter into LDS
    const int r = r0 + tid;
    if (r < B) {
        float x = x_in[r];
        float u = (x + 1.0f) * 32.0f;                            // span coordinate
        int s = (int)floorf(u);
        s = s < 0 ? 0 : (s > GRID_SIZE - 1 ? GRID_SIZE - 1 : s);
        int j = s + KDEG;                                        // knot span index
        float l1 = x - knotf(j),     l2 = x - knotf(j - 1), l3 = x - knotf(j - 2);
        float g1 = knotf(j + 1) - x, g2 = knotf(j + 2) - x, g3 = knotf(j + 3) - x;
        float N0 = 1.f, N1, N2, N3, saved, temp;
        // degree 1
        temp = N0 / (g1 + l1); N0 = g1 * temp;          N1 = l1 * temp;
        // degree 2
        temp = N0 / (g1 + l2); N0 = g1 * temp;          saved = l2 * temp;
        temp = N1 / (g2 + l1); N1 = saved + g2 * temp;  N2 = l1 * temp;
        // degree 3
        temp = N0 / (g1 + l3); N0 = g1 * temp;          saved = l3 * temp;
        temp = N1 / (g2 + l2); N1 = saved + g2 * temp;  saved = l2 * temp;
        temp = N2 / (g3 + l1); N2 = saved + g3 * temp;  N3 = l1 * temp;
        float* row = &tile[tid * NB + s];                        // columns s..s+3
        row[0] = N0; row[1] = N1; row[2] = N2; row[3] = N3;
    }
    __syncthreads();

    const int valid = (B - r0) < ROWS_PER_BLOCK ? (B - r0) : ROWS_PER_BLOCK;

#if __has_builtin(__builtin_amdgcn_tensor_store_from_lds)
    // 3a) TDM path: each wave DMAs its 32-row contiguous slice LDS -> HBM.
    //     TDM ignores EXEC, tracked by TENSORcnt; descriptor per ISA cdna5 ch.8.
    {
        // The TDM builtin only consumes integer descriptors, so the compiler
        // cannot see that the hardware reads the LDS tile. Without this
        // memory-clobbering asm that takes the tile address, every LDS store
        // above (and the whole compute chain) gets dead-store-eliminated.
        asm volatile("" : : "r"(tile) : "memory");

        int wave  = __builtin_amdgcn_readfirstlane(tid >> 5);    // SGPR-uniform
        int rowA  = wave * 32;
        int vs    = valid - rowA; vs = vs < 0 ? 0 : (vs > 32 ? 32 : vs);
        unsigned nelem   = (unsigned)(vs * NB);                  // <= 2144 elements
        unsigned lds_off = (unsigned)(rowA * NB * 4);            // tile at LDS offset 0
        unsigned long long gaddr =
            (unsigned long long)(out + (size_t)(r0 + rowA) * NB);
        v4u g0;
        g0[0] = 1u;                                   // count=1, user mode
        g0[1] = lds_off;                              // lds_addr (bytes)
        g0[2] = (unsigned)(gaddr & 0xFFFFFFFFull);    // global_addr[31:0]
        g0[3] = (unsigned)(gaddr >> 32) | (2u << 30); // global_addr[56:32] | type=2
        v8i g1;
        g1[0] = 0x20000;                              // data_size=2 -> 4-byte elems
        g1[1] = (int)(nelem << 16);                   // tensor_dim0[15:0]  @ bits 63:48
        g1[2] = (int)(nelem >> 16);                   // tensor_dim0[31:16] @ bits 79:64
        g1[3] = (int)((nelem & 0xFFFFu) << 16);       // tile_dim0          @ bits 127:112
        g1[4] = 0; g1[5] = 0; g1[6] = 0; g1[7] = 0;   // 1-D tile: dims 1/2 unused
        v4i g2 = {0, 0, 0, 0};
        v4i g3 = {0, 0, 0, 0};
  #if __clang_major__ >= 23
        v8i gx = {0, 0, 0, 0, 0, 0, 0, 0};
        __builtin_amdgcn_tensor_store_from_lds(g0, g1, g2, g3, gx, 0);
  #else
        __builtin_amdgcn_tensor_store_from_lds(g0, g1, g2, g3, 0);
  #endif
    }
  #if __has_builtin(__builtin_amdgcn_s_wait_tensorcnt)
    __builtin_amdgcn_s_wait_tensorcnt(0);
  #endif
#else
    // 3b) Fallback: coalesced b128 non-temporal stores (512 B contiguous / wave-instr)
    {
        const int nfloat = valid * NB;
        const int nf4    = nfloat >> 2;                // block base is 16B-aligned
        f4* out4 = (f4*)(out + (size_t)r0 * NB);
        for (int i = tid; i < nf4; i += NTHREADS)
            __builtin_nontemporal_store(tile4[i], &out4[i]);
        for (int i = (nf4 << 2) + tid; i < nfloat; i += NTHREADS)
            __builtin_nontemporal_store(tile[i], &out[(size_t)r0 * NB + i]);
    }
#endif
}

extern "C" void kernel_launch(void* const* d_in, const int* in_sizes, int n_in,
                              void* d_out, int out_size, void* d_ws, size_t ws_size,
                              hipStream_t stream) {
    const float* x = (const float*)d_in[0];
    float* out = (float*)d_out;
    const int B = in_sizes[0];
    const int nblocks = (B + ROWS_PER_BLOCK - 1) / ROWS_PER_BLOCK;
    spline_basis_kernel<<<nblocks, NTHREADS, 0, stream>>>(x, out, B);
}